// GNNActor_33243046871254
// MI455X (gfx1250) — compile-verified
//
#include <hip/hip_runtime.h>

#define N_NODES 50000
#define N_EDGES 600000
#define D_CH 128
#define H_CH 32

typedef float v2f __attribute__((ext_vector_type(2)));
typedef float v8f __attribute__((ext_vector_type(8)));

// ---------------- degree / normalization ----------------

__global__ void k_init_deg(float* __restrict__ deg) {
    int i = blockIdx.x * 256 + threadIdx.x;
    if (i < N_NODES) deg[i] = 1.0f;              // self-loop contributes 1
}

__global__ void k_count_deg(const int* __restrict__ ei, float* __restrict__ deg) {
    int e = blockIdx.x * 256 + threadIdx.x;
    if (e < N_EDGES) {
        int c = ei[N_EDGES + e];                 // target node
        unsafeAtomicAdd(&deg[c], 1.0f);          // global_atomic_add_f32
    }
}

__global__ void k_dinv(const float* __restrict__ deg, float* __restrict__ dinv) {
    int i = blockIdx.x * 256 + threadIdx.x;
    if (i < N_NODES) dinv[i] = rsqrtf(deg[i]);   // deg >= 1 always
}

// ---------------- xw = x @ W_gcn  (f32 WMMA, 16x16 tiles) ----------------
// A layout (16x4 f32): lane&15 = row M; VGPR0 = K=kb, VGPR1 = K=kb+1, kb = (lane>=16)?2:0
// B layout (4x16 f32): lane&15 = col N; VGPR0 = K=kb, VGPR1 = K=kb+1
// C/D (16x16 f32): VGPR j -> M = j + ((lane>=16)?8:0), N = lane&15

__global__ void __launch_bounds__(128) k_gemm_xw(const float* __restrict__ x,
                                                 const float* __restrict__ W,
                                                 float* __restrict__ xw) {
    int lane = threadIdx.x & 31;
    int wave = threadIdx.x >> 5;
    int tile = blockIdx.x * 4 + wave;            // 3125*8 = 25000 tiles, exact
    int rowTile = tile >> 3;
    int colTile = tile & 7;
    int rowi = lane & 15;
    int kb   = (lane >> 4) << 1;
    int col  = (colTile << 4) + rowi;
    const float* xr = x + (size_t)((rowTile << 4) + rowi) * D_CH;

    v8f acc = {};
    #pragma unroll
    for (int k = 0; k < D_CH; k += 4) {
        v2f a = *(const v2f*)(xr + k + kb);                  // 8B-aligned b64 load
        v2f b;
        b.x = W[(k + kb)     * D_CH + col];
        b.y = W[(k + kb + 1) * D_CH + col];
        acc = __builtin_amdgcn_wmma_f32_16x16x4_f32(false, a, false, b,
                                                    (short)0, acc, false, false);
    }
    int mbase = (rowTile << 4) + ((lane >> 4) << 3);
    #pragma unroll
    for (int j = 0; j < 8; ++j)
        xw[(size_t)(mbase + j) * D_CH + col] = acc[j];
}

// ---------------- acc init with self-loop term: acc = xw * dinv^2 ----------------

__global__ void k_self(const float* __restrict__ xw, const float* __restrict__ dinv,
                       float* __restrict__ acc) {
    int i = blockIdx.x * 256 + threadIdx.x;      // one float4 per thread
    if (i < N_NODES * D_CH / 4) {
        int node = i >> 5;                       // 32 float4 per node row
        float s = dinv[node];
        s = s * s;
        float4 v = ((const float4*)xw)[i];
        float4 o;
        o.x = v.x * s; o.y = v.y * s; o.z = v.z * s; o.w = v.w * s;
        ((float4*)acc)[i] = o;
    }
}

// ---------------- edge scatter: acc[col] += xw[row] * dinv[row]*dinv[col] ----------------
// one wave per edge; each lane handles 4 channels

__global__ void __launch_bounds__(256) k_scatter(const int* __restrict__ ei,
                                                 const float* __restrict__ xw,
                                                 const float* __restrict__ dinv,
                                                 float* __restrict__ acc) {
    int gid  = blockIdx.x * 256 + threadIdx.x;
    int e    = gid >> 5;
    int lane = gid & 31;
    if (e < N_EDGES) {
        int r = ei[e];
        int c = ei[N_EDGES + e];
        float nrm = dinv[r] * dinv[c];
        float4 v = *(const float4*)(xw + (size_t)r * D_CH + lane * 4);
        float* dst = acc + (size_t)c * D_CH + lane * 4;
        unsafeAtomicAdd(dst + 0, v.x * nrm);
        unsafeAtomicAdd(dst + 1, v.y * nrm);
        unsafeAtomicAdd(dst + 2, v.z * nrm);
        unsafeAtomicAdd(dst + 3, v.w * nrm);
    }
}

// ---------------- fused epilogue: relu(acc+b)+x -> MLP (WMMA) -> out ----------------
// one wave = 16 nodes; h0 (16x128) staged in LDS, h1/h2 re-staged for A-layout

__global__ void __launch_bounds__(128) k_mlp(const float* __restrict__ acc,
                                             const float* __restrict__ x,
                                             const float* __restrict__ bg,
                                             const float* __restrict__ w1,
                                             const float* __restrict__ b1,
                                             const float* __restrict__ w2,
                                             const float* __restrict__ b2,
                                             const float* __restrict__ w3,
                                             const float* __restrict__ b3,
                                             float* __restrict__ out) {
    __shared__ __align__(16) float lds[4][2048];  // 8KB per wave, 32KB per block
    int lane = threadIdx.x & 31;
    int wave = threadIdx.x >> 5;
    int tile = blockIdx.x * 4 + wave;
    bool valid = tile < (N_NODES / 16);          // 3125 tiles
    int t = valid ? tile : 0;
    float* sh = lds[wave];

    // phase 1: h0 = relu(acc + b_gcn) + x   into LDS (16 x 128)
    const float* accT = acc + (size_t)t * 16 * D_CH;
    const float* xT   = x   + (size_t)t * 16 * D_CH;
    #pragma unroll
    for (int i = 0; i < 16; ++i) {
        int off = (i * 32 + lane) * 4;
        float4 a  = *(const float4*)(accT + off);
        float4 xv = *(const float4*)(xT + off);
        float4 bgv = *(const float4*)(bg + (off & 127));
        float4 h;
        h.x = fmaxf(a.x + bgv.x, 0.f) + xv.x;
        h.y = fmaxf(a.y + bgv.y, 0.f) + xv.y;
        h.z = fmaxf(a.z + bgv.z, 0.f) + xv.z;
        h.w = fmaxf(a.w + bgv.w, 0.f) + xv.w;
        *(float4*)(sh + off) = h;
    }
    __syncthreads();

    int rowi = lane & 15;
    int kb   = (lane >> 4) << 1;
    int hi8  = (lane >> 4) << 3;

    // layer 1: h1 = relu(h0 @ w1 + b1)   [16x128]x[128x32]
    v8f h1v[2];
    #pragma unroll
    for (int c = 0; c < 2; ++c) {
        v8f a8 = {};
        int col = (c << 4) + rowi;
        #pragma unroll
        for (int k = 0; k < D_CH; k += 4) {
            v2f a = *(const v2f*)(sh + rowi * D_CH + k + kb);
            v2f b;
            b.x = w1[(k + kb)     * H_CH + col];
            b.y = w1[(k + kb + 1) * H_CH + col];
            a8 = __builtin_amdgcn_wmma_f32_16x16x4_f32(false, a, false, b,
                                                       (short)0, a8, false, false);
        }
        float bias = b1[col];
        #pragma unroll
        for (int j = 0; j < 8; ++j) h1v[c][j] = fmaxf(a8[j] + bias, 0.f);
    }
    __syncthreads();
    #pragma unroll
    for (int c = 0; c < 2; ++c)
        #pragma unroll
        for (int j = 0; j < 8; ++j)
            sh[(hi8 + j) * H_CH + (c << 4) + rowi] = h1v[c][j];   // h1 at [0,512)
    __syncthreads();

    // layer 2: h2 = relu(h1 @ w2 + b2)   [16x32]x[32x32]
    v8f h2v[2];
    #pragma unroll
    for (int c = 0; c < 2; ++c) {
        v8f a8 = {};
        int col = (c << 4) + rowi;
        #pragma unroll
        for (int k = 0; k < H_CH; k += 4) {
            v2f a = *(const v2f*)(sh + rowi * H_CH + k + kb);
            v2f b;
            b.x = w2[(k + kb)     * H_CH + col];
            b.y = w2[(k + kb + 1) * H_CH + col];
            a8 = __builtin_amdgcn_wmma_f32_16x16x4_f32(false, a, false, b,
                                                       (short)0, a8, false, false);
        }
        float bias = b2[col];
        #pragma unroll
        for (int j = 0; j < 8; ++j) h2v[c][j] = fmaxf(a8[j] + bias, 0.f);
    }
    __syncthreads();
    #pragma unroll
    for (int c = 0; c < 2; ++c)
        #pragma unroll
        for (int j = 0; j < 8; ++j)
            sh[1024 + (hi8 + j) * H_CH + (c << 4) + rowi] = h2v[c][j];  // h2 at [1024,1536)
    __syncthreads();

    // layer 3: out = h2 @ w3 + b3   (matvec, lanes 0..15 = one node each)
    if (lane < 16) {
        float s = b3[0];
        #pragma unroll
        for (int k = 0; k < H_CH; ++k)
            s += sh[1024 + lane * H_CH + k] * w3[k];
        if (valid) out[t * 16 + lane] = s;
    }
}

// ---------------- launch ----------------

extern "C" void kernel_launch(void* const* d_in, const int* in_sizes, int n_in,
                              void* d_out, int out_size, void* d_ws, size_t ws_size,
                              hipStream_t stream) {
    const float* x  = (const float*)d_in[0];
    const int*   ei = (const int*)d_in[1];
    const float* Wg = (const float*)d_in[2];
    const float* bg = (const float*)d_in[3];
    const float* w1 = (const float*)d_in[4];
    const float* b1 = (const float*)d_in[5];
    const float* w2 = (const float*)d_in[6];
    const float* b2 = (const float*)d_in[7];
    const float* w3 = (const float*)d_in[8];
    const float* b3 = (const float*)d_in[9];
    float* out = (float*)d_out;

    float* ws   = (float*)d_ws;
    float* xw   = ws;                                  // N*D floats
    float* acc  = xw  + (size_t)N_NODES * D_CH;        // N*D floats
    float* deg  = acc + (size_t)N_NODES * D_CH;        // N floats
    float* dinv = deg + N_NODES;                       // N floats

    k_init_deg <<<(N_NODES + 255) / 256, 256, 0, stream>>>(deg);
    k_count_deg<<<(N_EDGES + 255) / 256, 256, 0, stream>>>(ei, deg);
    k_dinv     <<<(N_NODES + 255) / 256, 256, 0, stream>>>(deg, dinv);
    k_gemm_xw  <<<25000 / 4, 128, 0, stream>>>(x, Wg, xw);
    k_self     <<<(N_NODES * D_CH / 4 + 255) / 256, 256, 0, stream>>>(xw, dinv, acc);
    k_scatter  <<<(N_EDGES * 32) / 256, 256, 0, stream>>>(ei, xw, dinv, acc);
    k_mlp      <<<(3125 + 3) / 4, 128, 0, stream>>>(acc, x, bg, w1, b1, w2, b2, w3, b3, out);
}